// SimpleGATLayer_18854906429657
// MI455X (gfx1250) — compile-verified
//
#include <hip/hip_runtime.h>
#include <hip/hip_bf16.h>
#include <math.h>

// ---------------------------------------------------------------------------
// GAT layer for gfx1250 (MI455X), wave32.
//  K1: h = x @ W            -- v_wmma_f32_16x16x4_f32, W staged in LDS
//  K2: init out/max/denom
//  K3: s_src/s_dst per node
//  K4: segment max  (native f32 atomic via int-punning max/min)
//  K5: segment sum of exp   (global_atomic_add_f32)
//  K6: alpha-weighted scatter-add of h[src] into out[dst]
// All irregular traffic stays inside the 192 MB L2 (h+out+scores < 120 MB).
// ---------------------------------------------------------------------------

typedef __attribute__((ext_vector_type(2))) float v2f;
typedef __attribute__((ext_vector_type(8))) float v8f;

#define IN_DIM   128
#define HEADS    4
#define OUT_DIM  32
#define HD       128          // HEADS*OUT_DIM
#define NEG_SLOPE 0.2f
#define LDSW     132          // padded row stride for W in LDS (bank-conflict free)

// ---------------- K1: GEMM h = x @ W via WMMA f32 16x16x4 ------------------
__global__ __launch_bounds__(256) void gat_gemm_wmma(
    const float* __restrict__ x, const float* __restrict__ W,
    float* __restrict__ h, int n_nodes)
{
    __shared__ float wlds[IN_DIM * LDSW];   // 128 x 132 floats = 67.6 KB

    // cooperative load of W (128x128) into padded LDS
    for (int i = threadIdx.x; i < IN_DIM * HD; i += 256) {
        int r = i >> 7, c = i & 127;
        wlds[r * LDSW + c] = W[i];
    }
    __syncthreads();

    const int wave = threadIdx.x >> 5;
    const int lane = threadIdx.x & 31;
    const int m0   = (blockIdx.x * 8 + wave) * 16;   // 16 rows per wave
    if (m0 >= n_nodes) return;                       // wave-uniform: EXEC stays all-1s

    // A-fragment addressing (16x4 f32): lanes 0-15 -> K=0,1 ; lanes 16-31 -> K=2,3
    const int mrow  = m0 + (lane & 15);
    const int khalf = (lane >> 4) << 1;              // 0 or 2
    const float* xrow = x + (size_t)mrow * IN_DIM;

    v8f acc[8] = {};                                 // 8 N-tiles of 16 cols -> 128 cols

    #pragma unroll 4
    for (int k = 0; k < IN_DIM; k += 4) {
        v2f a;
        a.x = xrow[k + khalf];
        a.y = xrow[k + khalf + 1];
        #pragma unroll
        for (int t = 0; t < 8; ++t) {
            const int col = t * 16 + (lane & 15);
            v2f b;
            b.x = wlds[(k + khalf)     * LDSW + col];
            b.y = wlds[(k + khalf + 1) * LDSW + col];
            // 8 args: (neg_a, A, neg_b, B, c_mod, C, reuse_a, reuse_b)
            acc[t] = __builtin_amdgcn_wmma_f32_16x16x4_f32(
                false, a, false, b, (short)0, acc[t], false, false);
        }
    }

    // C/D layout: lane<16 -> rows m0+v ; lane>=16 -> rows m0+8+v ; col fixed/lane
    const int rbase = m0 + ((lane >> 4) << 3);
    const int cbase = lane & 15;
    #pragma unroll
    for (int t = 0; t < 8; ++t)
        #pragma unroll
        for (int v = 0; v < 8; ++v)
            h[(size_t)(rbase + v) * HD + t * 16 + cbase] = acc[t][v];
}

// ---------------- K2: init out / maxbuf / denom ----------------------------
__global__ __launch_bounds__(256) void gat_init(
    float* __restrict__ out, float* __restrict__ maxbuf,
    float* __restrict__ denom, int n_nodes)
{
    int i = blockIdx.x * 256 + threadIdx.x;
    if (i < n_nodes * HD) out[i] = 0.0f;
    if (i < n_nodes * HEADS) { maxbuf[i] = -__builtin_inff(); denom[i] = 0.0f; }
}

// ---------------- K3: per-node score halves --------------------------------
__global__ __launch_bounds__(256) void gat_scores(
    const float* __restrict__ h, const float* __restrict__ a_src,
    const float* __restrict__ a_dst, float* __restrict__ s_src,
    float* __restrict__ s_dst, int n_nodes)
{
    int idx = blockIdx.x * 256 + threadIdx.x;        // node*HEADS + head
    if (idx >= n_nodes * HEADS) return;
    const int head = idx & (HEADS - 1);
    const float* hp = h + (size_t)(idx >> 2) * HD + head * OUT_DIM;
    const float* as = a_src + head * OUT_DIM;
    const float* ad = a_dst + head * OUT_DIM;
    float ss = 0.f, sd = 0.f;
    #pragma unroll
    for (int i = 0; i < OUT_DIM; ++i) { float v = hp[i]; ss += v * as[i]; sd += v * ad[i]; }
    s_src[idx] = ss;
    s_dst[idx] = sd;
}

// float atomic max via signed/unsigned int atomics (exact, single HW atomic)
__device__ __forceinline__ void atomicMaxF(float* addr, float val) {
    if (val >= 0.0f) atomicMax((int*)addr, __float_as_int(val));
    else             atomicMin((unsigned int*)addr, __float_as_uint(val));
}

__device__ __forceinline__ float leaky(float v) {
    return v > 0.0f ? v : NEG_SLOPE * v;
}

// ---------------- K4: segment max over dst ---------------------------------
__global__ __launch_bounds__(256) void gat_edge_max(
    const int* __restrict__ ei, const float* __restrict__ s_src,
    const float* __restrict__ s_dst, float* __restrict__ maxbuf, int n_edges)
{
    int e = blockIdx.x * 256 + threadIdx.x;
    if (e >= n_edges) return;
    const int s = ei[e], d = ei[n_edges + e];
    #pragma unroll
    for (int hd = 0; hd < HEADS; ++hd)
        atomicMaxF(&maxbuf[d * HEADS + hd],
                   leaky(s_src[s * HEADS + hd] + s_dst[d * HEADS + hd]));
}

// ---------------- K5: segment sum of exp -----------------------------------
__global__ __launch_bounds__(256) void gat_edge_denom(
    const int* __restrict__ ei, const float* __restrict__ s_src,
    const float* __restrict__ s_dst, const float* __restrict__ maxbuf,
    float* __restrict__ denom, int n_edges)
{
    int e = blockIdx.x * 256 + threadIdx.x;
    if (e >= n_edges) return;
    const int s = ei[e], d = ei[n_edges + e];
    #pragma unroll
    for (int hd = 0; hd < HEADS; ++hd) {
        float v  = leaky(s_src[s * HEADS + hd] + s_dst[d * HEADS + hd]);
        float ex = __expf(v - maxbuf[d * HEADS + hd]);
        unsafeAtomicAdd(&denom[d * HEADS + hd], ex);   // global_atomic_add_f32
    }
}

// ---------------- K6: alpha-weighted scatter-add ---------------------------
// one wave per edge: 32 lanes x float4 = 128 floats
__global__ __launch_bounds__(256) void gat_edge_aggr(
    const int* __restrict__ ei, const float* __restrict__ h,
    const float* __restrict__ s_src, const float* __restrict__ s_dst,
    const float* __restrict__ maxbuf, const float* __restrict__ denom,
    float* __restrict__ out, int n_edges)
{
    const int e = blockIdx.x * 8 + (threadIdx.x >> 5);
    if (e >= n_edges) return;
    const int lane = threadIdx.x & 31;
    const int s = ei[e], d = ei[n_edges + e];
    const int hd = lane >> 3;                        // head owning this lane's 4 floats

    float v = leaky(s_src[s * HEADS + hd] + s_dst[d * HEADS + hd]);
    float alpha = __expf(v - maxbuf[d * HEADS + hd]) /
                  (denom[d * HEADS + hd] + 1e-9f);

    const float4 hv = ((const float4*)(h + (size_t)s * HD))[lane];
    float* op = out + (size_t)d * HD + lane * 4;
    unsafeAtomicAdd(op + 0, hv.x * alpha);
    unsafeAtomicAdd(op + 1, hv.y * alpha);
    unsafeAtomicAdd(op + 2, hv.z * alpha);
    unsafeAtomicAdd(op + 3, hv.w * alpha);
}

// ---------------------------------------------------------------------------
extern "C" void kernel_launch(void* const* d_in, const int* in_sizes, int n_in,
                              void* d_out, int out_size, void* d_ws, size_t ws_size,
                              hipStream_t stream)
{
    const float* x     = (const float*)d_in[0];
    const int*   ei    = (const int*)  d_in[1];
    const float* W     = (const float*)d_in[2];
    const float* a_src = (const float*)d_in[3];
    const float* a_dst = (const float*)d_in[4];
    float* out = (float*)d_out;

    const int n_nodes = in_sizes[0] / IN_DIM;
    const int n_edges = in_sizes[1] / 2;

    // workspace layout: h | s_src | s_dst | maxbuf | denom  (~57.7 MB)
    float* h      = (float*)d_ws;
    float* s_src  = h     + (size_t)n_nodes * HD;
    float* s_dst  = s_src + (size_t)n_nodes * HEADS;
    float* maxbuf = s_dst + (size_t)n_nodes * HEADS;
    float* denom  = maxbuf+ (size_t)n_nodes * HEADS;

    const int gemm_blocks = (n_nodes + 127) / 128;          // 128 rows / block (8 waves)
    gat_gemm_wmma<<<gemm_blocks, 256, 0, stream>>>(x, W, h, n_nodes);

    gat_init<<<(n_nodes * HD + 255) / 256, 256, 0, stream>>>(out, maxbuf, denom, n_nodes);

    gat_scores<<<(n_nodes * HEADS + 255) / 256, 256, 0, stream>>>(
        h, a_src, a_dst, s_src, s_dst, n_nodes);

    gat_edge_max<<<(n_edges + 255) / 256, 256, 0, stream>>>(
        ei, s_src, s_dst, maxbuf, n_edges);

    gat_edge_denom<<<(n_edges + 255) / 256, 256, 0, stream>>>(
        ei, s_src, s_dst, maxbuf, denom, n_edges);

    gat_edge_aggr<<<(n_edges + 7) / 8, 256, 0, stream>>>(
        ei, h, s_src, s_dst, maxbuf, denom, out, n_edges);
}